// Model_91036126806550
// MI455X (gfx1250) — compile-verified
//
#include <hip/hip_runtime.h>
#include <stdint.h>

#define BLOCK 256
#define GROUPS_PER_BLOCK 256
#define PACKED_DW 17                                   // dwords per group (16 data + 1 scale)
#define CHUNK_DW (GROUPS_PER_BLOCK * PACKED_DW)        // 4352 dwords per block
#define CHUNK_B16 (CHUNK_DW / 4)                       // 1088 16-byte units (17408 B, 16B aligned)

typedef float v4f __attribute__((ext_vector_type(4)));

// Dequant one group given its 17 dwords; writes 8 non-temporal float4 (32 floats).
__device__ __forceinline__ void dequant_group(const uint32_t* __restrict__ gp,
                                              float* __restrict__ outp) {
  uint32_t su = gp[16];                                 // 0..255 shared exponent byte
  uint32_t eb = su < 1u ? 1u : (su > 254u ? 254u : su); // == clip(su-127,-126,127)+127
  float s = __uint_as_float(eb << 23);                  // 2^log_scale, exact
  float s12 = s * (1.0f / 12.0f);                       // power-of-2 scale of RN(1/12)
  v4f* ov = reinterpret_cast<v4f*>(outp);
#pragma unroll
  for (int k = 0; k < 8; ++k) {
    uint32_t d0 = gp[2 * k];
    uint32_t d1 = gp[2 * k + 1];
    v4f r;
    r.x = (float)((int)(d0 & 15u) - 8) * s12;
    r.y = (float)((int)((d0 >> 4) & 15u) - 8) * s12;
    r.z = (float)((int)(d1 & 15u) - 8) * s12;
    r.w = (float)((int)((d1 >> 4) & 15u) - 8) * s12;
    __builtin_nontemporal_store(r, &ov[k]);            // global_store_b128 th:TH_STORE_NT
  }
}

__global__ __launch_bounds__(BLOCK) void mxfp4_dequant_async(
    const uint32_t* __restrict__ packed, float* __restrict__ out) {
  __shared__ uint32_t lds_in[CHUNK_DW];

  const int tid = threadIdx.x;
  const uint64_t in_base =
      (uint64_t)(uintptr_t)packed + (uint64_t)blockIdx.x * (uint64_t)(CHUNK_DW * 4);
  // Workgroup-relative LDS byte address = low 32 bits of flat address.
  const uint32_t lds_base = (uint32_t)(uintptr_t)(void*)lds_in;

  // --- Stage 1: async bulk copy of this block's packed chunk into LDS (ASYNCcnt path).
  // Stream is read exactly once -> non-temporal so it doesn't churn the 192MB L2.
  // 1088 x 16B transfers; 4 per thread + 64 extra on the first two waves.
#pragma unroll
  for (int k = 0; k < 4; ++k) {
    const int j = tid + k * BLOCK;
    const uint64_t ga = in_base + (uint64_t)j * 16u;
    const uint32_t la = lds_base + (uint32_t)j * 16u;
    asm volatile("global_load_async_to_lds_b128 %0, %1, off th:TH_LOAD_NT"
                 :: "v"(la), "v"(ga) : "memory");
  }
  if (tid < (CHUNK_B16 - 4 * BLOCK)) {  // 64 remaining 16B units
    const int j = 4 * BLOCK + tid;
    const uint64_t ga = in_base + (uint64_t)j * 16u;
    const uint32_t la = lds_base + (uint32_t)j * 16u;
    asm volatile("global_load_async_to_lds_b128 %0, %1, off th:TH_LOAD_NT"
                 :: "v"(la), "v"(ga) : "memory");
  }
  asm volatile("s_wait_asynccnt 0" ::: "memory");
  __syncthreads();

  // --- Stage 2: thread t dequantizes group t of this block.
  // LDS read stride 17 dwords is coprime with 64 banks -> conflict-free.
  const uint32_t* gp = &lds_in[tid * PACKED_DW];
  const size_t g = (size_t)blockIdx.x * GROUPS_PER_BLOCK + (size_t)tid;
  dequant_group(gp, out + g * 32);
}

// Scalar tail for num_groups % GROUPS_PER_BLOCK (0 for the reference shape).
__global__ __launch_bounds__(BLOCK) void mxfp4_dequant_tail(
    const uint32_t* __restrict__ packed, float* __restrict__ out,
    int start_group, int num_groups) {
  const int g = start_group + (int)(blockIdx.x * blockDim.x + threadIdx.x);
  if (g >= num_groups) return;
  uint32_t buf[PACKED_DW];
  const uint32_t* gp = packed + (size_t)g * PACKED_DW;
#pragma unroll
  for (int k = 0; k < PACKED_DW; ++k) buf[k] = gp[k];
  dequant_group(buf, out + (size_t)g * 32);
}

extern "C" void kernel_launch(void* const* d_in, const int* in_sizes, int n_in,
                              void* d_out, int out_size, void* d_ws, size_t ws_size,
                              hipStream_t stream) {
  (void)n_in; (void)out_size; (void)d_ws; (void)ws_size;
  const uint32_t* packed = (const uint32_t*)d_in[0];
  float* out = (float*)d_out;

  const int num_groups = in_sizes[0] / PACKED_DW;      // 4,194,304 for reference shape
  const int full_blocks = num_groups / GROUPS_PER_BLOCK;

  if (full_blocks > 0) {
    hipLaunchKernelGGL(mxfp4_dequant_async, dim3(full_blocks), dim3(BLOCK), 0, stream,
                       packed, out);
  }
  const int rem = num_groups - full_blocks * GROUPS_PER_BLOCK;
  if (rem > 0) {
    const int start = full_blocks * GROUPS_PER_BLOCK;
    const int tb = (rem + BLOCK - 1) / BLOCK;
    hipLaunchKernelGGL(mxfp4_dequant_tail, dim3(tb), dim3(BLOCK), 0, stream,
                       packed, out, start, num_groups);
  }
}